// AssignYolo_55697135894736
// MI455X (gfx1250) — compile-verified
//
#include <hip/hip_runtime.h>
#include <hip/hip_bf16.h>
#include <stdint.h>

#define THRESH 0.3f
#define MAX_GT 128
#define COL_SPLIT 4

// ---------------------------------------------------------------------------
// CDNA5 async-to-LDS: per-lane DMA of 16B global -> LDS, ASYNCcnt-tracked.
// ---------------------------------------------------------------------------
__device__ __forceinline__ void async_load_lds_b128(void* lds_ptr, const void* gptr) {
#if defined(__gfx1250__)
    unsigned lds_off = (unsigned)(uintptr_t)lds_ptr;
    asm volatile("global_load_async_to_lds_b128 %0, %1, off"
                 :
                 : "v"(lds_off), "v"(gptr)
                 : "memory");
#else
    *(float4*)lds_ptr = *(const float4*)gptr;
#endif
}

__device__ __forceinline__ void wait_asynccnt0() {
#if defined(__gfx1250__)
#if __has_builtin(__builtin_amdgcn_s_wait_asynccnt)
    __builtin_amdgcn_s_wait_asynccnt(0);
#else
    asm volatile("s_wait_asynccnt 0x0" ::: "memory");
#endif
#endif
}

// ---------------------------------------------------------------------------
// Kernel 1: row-wise. One thread per anchor; GT table async-DMA'd into LDS.
// Division-free threshold: iou >= T  <=>  inter*(1+T) >= T*areaA + T*areaG.
// ---------------------------------------------------------------------------
__global__ void __launch_bounds__(256)
row_assign_kernel(const float4* __restrict__ anchor,
                  const float4* __restrict__ gt,
                  int* __restrict__ assign,
                  int nA, int nG) {
    __shared__ float4 s_gt[MAX_GT];
    __shared__ float  s_tg[MAX_GT];   // THRESH * areaG

    const int t = threadIdx.x;

    if (t < nG) {
        async_load_lds_b128(&s_gt[t], &gt[t]);
    }
    wait_asynccnt0();
    __syncthreads();

    if (t < nG) {
        float4 g = s_gt[t];
        s_tg[t] = THRESH * (g.z - g.x) * (g.w - g.y);
    }
    __syncthreads();

    const int n = blockIdx.x * blockDim.x + t;
    if (n >= nA) return;

    const float4 a  = anchor[n];
    const float  ta = THRESH * (a.z - a.x) * (a.w - a.y);

    int any = 0;
#pragma unroll 8
    for (int j = 0; j < nG; ++j) {
        float4 g = s_gt[j];
        float lx = fmaxf(a.x, g.x);
        float ly = fmaxf(a.y, g.y);
        float rx = fminf(a.z, g.z);
        float ry = fminf(a.w, g.w);
        float w  = fmaxf(rx - lx, 0.0f);
        float h  = fmaxf(ry - ly, 0.0f);
        float inter = w * h;
        any |= (inter * (1.0f + THRESH) >= ta + s_tg[j]) ? 1 : 0;
    }
    assign[n] = any ? -2 : -1;
}

// ---------------------------------------------------------------------------
// Shared device helper: block-level max-reduce of a packed 64-bit key
// (iou_bits<<32 | ~idx) via wave32 shuffles + one LDS pass.
// ---------------------------------------------------------------------------
__device__ __forceinline__ unsigned long long
block_reduce_max_key(unsigned long long key, int t) {
    for (int o = 16; o > 0; o >>= 1) {
        unsigned long long other = __shfl_down(key, o);
        if (other > key) key = other;
    }
    __shared__ unsigned long long s_keys[32];
    const int lane = t & 31;
    const int wid  = t >> 5;
    if (lane == 0) s_keys[wid] = key;
    __syncthreads();
    key = 0ull;
    if (wid == 0) {
        key = s_keys[lane];
        for (int o = 16; o > 0; o >>= 1) {
            unsigned long long other = __shfl_down(key, o);
            if (other > key) key = other;
        }
    }
    return key;   // valid in (wid==0, lane==0)
}

// ---------------------------------------------------------------------------
// Per-thread scan core: division-free best tracking via cross-multiplication
// (inter_c/uni_c > binter/buni  <=>  inter_c*buni > binter*uni_c, all > 0).
// One divide per THREAD at the end, instead of one per anchor.
// ---------------------------------------------------------------------------
__device__ __forceinline__ unsigned long long
scan_best_key(const float4* __restrict__ anchor, float4 g, float areaG,
              int start, int end, int stride, int t) {
    float binter = -1.0f, buni = 1.0f;
    int   bidx = -1;

#pragma unroll 4
    for (int n = start + t; n < end; n += stride) {
#if defined(__gfx1250__)
        __builtin_prefetch(&anchor[n + stride], 0, 1);   // global_prefetch_b8
#endif
        float4 a = anchor[n];
        float areaA = (a.z - a.x) * (a.w - a.y);
        float lx = fmaxf(a.x, g.x);
        float ly = fmaxf(a.y, g.y);
        float rx = fminf(a.z, g.z);
        float ry = fminf(a.w, g.w);
        float w  = fmaxf(rx - lx, 0.0f);
        float h  = fmaxf(ry - ly, 0.0f);
        float inter = w * h;
        float uni   = areaA + areaG - inter;
        if (inter * buni > binter * uni) {               // strict > keeps first max
            binter = inter; buni = uni; bidx = n;
        }
    }
    if (bidx < 0) return 0ull;
    float biou = binter / buni;                          // single divide per thread
    return ((unsigned long long)__float_as_uint(biou) << 32) |
           (unsigned)(~(unsigned)bidx);
}

// ---------------------------------------------------------------------------
// Kernel 2a (split path): COL_SPLIT blocks per GT, partial argmax merged into
// workspace via 64-bit global atomicMax on the packed key.
// ---------------------------------------------------------------------------
__global__ void __launch_bounds__(1024)
col_partial_kernel(const float4* __restrict__ anchor,
                   const float4* __restrict__ gt,
                   unsigned long long* __restrict__ ws_keys,
                   int nA) {
    const int m    = blockIdx.x >> 2;                   // COL_SPLIT == 4
    const int part = blockIdx.x & (COL_SPLIT - 1);
    const int t    = threadIdx.x;

    const float4 g = gt[m];
    const float areaG = (g.z - g.x) * (g.w - g.y);

    const int chunk = (nA + COL_SPLIT - 1) / COL_SPLIT;
    const int start = part * chunk;
    const int end   = min(start + chunk, nA);

    unsigned long long key =
        scan_best_key(anchor, g, areaG, start, end, blockDim.x, t);
    key = block_reduce_max_key(key, t);
    if (t == 0 && key != 0ull) {
        atomicMax(&ws_keys[m], key);
    }
}

__global__ void __launch_bounds__(MAX_GT)
ws_init_kernel(unsigned long long* __restrict__ ws_keys, int nG) {
    int t = threadIdx.x;
    if (t < nG) ws_keys[t] = 0ull;
}

__global__ void __launch_bounds__(MAX_GT)
final_assign_kernel(const unsigned long long* __restrict__ ws_keys,
                    int* __restrict__ assign, int nG) {
    int m = threadIdx.x;
    if (m < nG) {
        unsigned long long key = ws_keys[m];
        int idx = (int)(~(unsigned)(key & 0xffffffffull));
        atomicMax(&assign[idx], m);                      // .at[].max: max gt id wins
    }
}

// ---------------------------------------------------------------------------
// Kernel 2b (fallback, no workspace): one block per GT does the full scan.
// ---------------------------------------------------------------------------
__global__ void __launch_bounds__(1024)
col_argmax_kernel(const float4* __restrict__ anchor,
                  const float4* __restrict__ gt,
                  int* __restrict__ assign,
                  int nA) {
    const int m = blockIdx.x;
    const int t = threadIdx.x;

    const float4 g = gt[m];
    const float areaG = (g.z - g.x) * (g.w - g.y);

    unsigned long long key =
        scan_best_key(anchor, g, areaG, 0, nA, blockDim.x, t);
    key = block_reduce_max_key(key, t);
    if (t == 0) {
        int idx = (int)(~(unsigned)(key & 0xffffffffull));
        atomicMax(&assign[idx], m);
    }
}

// ---------------------------------------------------------------------------
extern "C" void kernel_launch(void* const* d_in, const int* in_sizes, int n_in,
                              void* d_out, int out_size, void* d_ws, size_t ws_size,
                              hipStream_t stream) {
    (void)n_in; (void)out_size;

    const float4* anchor = (const float4*)d_in[0];    // [N,4] f32
    const float4* gt     = (const float4*)d_in[1];    // [M,4] f32
    int*          assign = (int*)d_out;               // [N] int32

    const int nA = in_sizes[0] / 4;
    const int nG = in_sizes[1] / 4;                   // 128

    // Pass 1: row threshold -> assign in {-1,-2} for every anchor.
    row_assign_kernel<<<dim3((nA + 255) / 256), dim3(256), 0, stream>>>(
        anchor, gt, assign, nA, nG);

    if (d_ws != nullptr && ws_size >= (size_t)nG * sizeof(unsigned long long)) {
        // Pass 2 (split): 4 blocks per GT -> 512 blocks, better WGP coverage.
        unsigned long long* ws_keys = (unsigned long long*)d_ws;
        ws_init_kernel<<<dim3(1), dim3(MAX_GT), 0, stream>>>(ws_keys, nG);
        col_partial_kernel<<<dim3(nG * COL_SPLIT), dim3(1024), 0, stream>>>(
            anchor, gt, ws_keys, nA);
        final_assign_kernel<<<dim3(1), dim3(MAX_GT), 0, stream>>>(
            ws_keys, assign, nG);
    } else {
        // Fallback: one block per GT, direct atomicMax into assign.
        col_argmax_kernel<<<dim3(nG), dim3(1024), 0, stream>>>(
            anchor, gt, assign, nA);
    }
}